// EncoderExact1D_5342939316844
// MI455X (gfx1250) — compile-verified
//
#include <hip/hip_runtime.h>

// Elementwise quantizer: out = clamp(trunc(x * 1024), 0, 1023) * 2^-10
// Pure HBM-streaming workload: 8 bytes/element, ~268 MB total -> bandwidth bound.
// Strategy: B128 loads/stores, non-temporal hints (footprint > 192MB L2, single touch),
// 4x float4 per thread for memory-level parallelism, wave32-friendly 256-thread blocks.

typedef float v4f __attribute__((ext_vector_type(4)));

#define TPB 256          // 8 waves per block on wave32
#define V4_PER_THREAD 4  // 4 x b128 in flight per lane

__device__ __forceinline__ float quant1(float x) {
    // x / 2^-10 == x * 1024.0f exactly (power-of-two scale)
    float t = truncf(x * 1024.0f);
    t = fmaxf(t, 0.0f);
    t = fminf(t, 1023.0f);
    return t * (1.0f / 1024.0f);
}

__global__ __launch_bounds__(TPB) void quantize_v4_kernel(const v4f* __restrict__ in,
                                                          v4f* __restrict__ out,
                                                          int n4) {
    const int base = blockIdx.x * (TPB * V4_PER_THREAD) + threadIdx.x;
#pragma unroll
    for (int k = 0; k < V4_PER_THREAD; ++k) {
        const int i = base + k * TPB;
        if (i < n4) {
            v4f v = __builtin_nontemporal_load(&in[i]);  // global_load_b128 th:NT
            v4f r;
            r.x = quant1(v.x);
            r.y = quant1(v.y);
            r.z = quant1(v.z);
            r.w = quant1(v.w);
            __builtin_nontemporal_store(r, &out[i]);     // global_store_b128 th:NT
        }
    }
}

// Scalar tail for n % 4 != 0 (not hit for N = 2^25, kept for generality).
__global__ __launch_bounds__(64) void quantize_tail_kernel(const float* __restrict__ in,
                                                           float* __restrict__ out,
                                                           int start, int n) {
    const int i = start + blockIdx.x * 64 + threadIdx.x;
    if (i < n) {
        out[i] = quant1(in[i]);
    }
}

extern "C" void kernel_launch(void* const* d_in, const int* in_sizes, int n_in,
                              void* d_out, int out_size, void* d_ws, size_t ws_size,
                              hipStream_t stream) {
    (void)n_in; (void)d_ws; (void)ws_size; (void)out_size;
    const float* x = (const float*)d_in[0];
    float* out = (float*)d_out;
    const int n  = in_sizes[0];
    const int n4 = n >> 2;  // number of float4 packets

    if (n4 > 0) {
        const int v4_per_block = TPB * V4_PER_THREAD;
        const int blocks = (n4 + v4_per_block - 1) / v4_per_block;
        quantize_v4_kernel<<<blocks, TPB, 0, stream>>>((const v4f*)x, (v4f*)out, n4);
    }
    const int done = n4 << 2;
    const int rem  = n - done;
    if (rem > 0) {
        const int blocks = (rem + 63) / 64;
        quantize_tail_kernel<<<blocks, 64, 0, stream>>>(x, out, done, n);
    }
}